// Fold3D_24824910970985
// MI455X (gfx1250) — compile-verified
//
#include <hip/hip_runtime.h>
#include <stdint.h>

// Fold3D (stride == kernel) = pure permutation, HBM-bandwidth bound:
// 512 MiB total traffic -> ~23 us floor at 23.3 TB/s on MI455X.
// Strategy: per-tile (b,c,p1,e1) 64x64 f32 transpose (p3 <-> e2) staged
// through LDS using the CDNA5 async global->LDS DMA path (ASYNCcnt),
// so loads are 256B-contiguous bursts and stores are a fully contiguous
// 16 KB block per workgroup.

#define TILE_ROWS   64          // p2*8 + p3
#define ROW_F4      16          // 64 floats per row = 16 float4
#define LDS_PITCH   72          // floats; padded (64+8) for bank spread
#define ROW_STRIDE  32768       // floats between input rows (one patch = 64ch*512vox)

__global__ __launch_bounds__(256) void fold3d_tile_kernel(
    const float* __restrict__ in, float* __restrict__ out) {
  __shared__ float lds[TILE_ROWS * LDS_PITCH];   // 18 KB

  const int tid  = threadIdx.x;
  const int tile = blockIdx.x;          // ((b*64 + c)*8 + p1)*8 + e1
  const int e1 = tile & 7;
  const int p1 = (tile >> 3) & 7;
  const int c  = (tile >> 6) & 63;
  const int b  = tile >> 12;

  // Input tile base (floats): x[b, p1, p2, p3, c, e1, e2, e3]
  const size_t in_base = (size_t)(b * 512 + p1 * 64) * (size_t)ROW_STRIDE
                       + (size_t)(c * 512 + e1 * 64);
  const float* gbase = in + in_base;

  // ---- Phase 1: async DMA 64 rows x 256B from global into LDS ------------
  // Each lane issues one b128 async copy; a wave covers two full 256B rows.
#pragma unroll
  for (int i = 0; i < 4; ++i) {
    const int idx = tid + i * 256;          // 0..1023 float4 slots
    const int row = idx >> 4;               // 0..63  (= p2*8 + p3)
    const int col = idx & 15;               // float4 within row (e2*2 + e3/4)
    const float* g = gbase + (size_t)row * ROW_STRIDE + (size_t)(col * 4);
    const unsigned lds_off =
        (unsigned)(uintptr_t)&lds[row * LDS_PITCH + col * 4];  // LDS byte addr
    asm volatile("global_load_async_to_lds_b128 %0, %1, off"
                 :: "v"(lds_off), "v"(g)
                 : "memory");
  }
  asm volatile("s_wait_asynccnt 0" ::: "memory");
  __syncthreads();

  // ---- Phase 2: transposed LDS read, fully contiguous 16 KB tile store ---
  // out tile element = (p2*8+e2)*64 + p3*8 + e3 ; lds[(p2*8+p3)][e2*8+e3]
  float4* outp = (float4*)(out + (size_t)tile * 4096);
#pragma unroll
  for (int i = 0; i < 4; ++i) {
    const int q   = tid + i * 256;          // output float4 index 0..1023
    const int r   = q >> 4;                 // p2*8 + e2
    const int rem = q & 15;
    const int p3  = rem >> 1;
    const int e3h = (rem & 1) * 4;
    const int p2  = r >> 3;
    const int e2  = r & 7;
    const float* l = &lds[(p2 * 8 + p3) * LDS_PITCH + e2 * 8 + e3h];
    outp[q] = *(const float4*)l;            // global_store_b128, 512B/wave
  }
}

extern "C" void kernel_launch(void* const* d_in, const int* in_sizes, int n_in,
                              void* d_out, int out_size, void* d_ws, size_t ws_size,
                              hipStream_t stream) {
  const float* x = (const float*)d_in[0];
  float* out = (float*)d_out;
  // elements per batch = 512 patches * 64 ch * 512 vox = 2^24
  const int B = in_sizes[0] >> 24;
  const int tiles = B * 64 /*c*/ * 8 /*p1*/ * 8 /*e1*/;
  fold3d_tile_kernel<<<dim3(tiles), dim3(256), 0, stream>>>(x, out);
}